// MiniBatchDiscrimination_19593640804692
// MI455X (gfx1250) — compile-verified
//
#include <hip/hip_runtime.h>
#include <hip/hip_bf16.h>

#define B_DIM   256
#define IN_F    512
#define OUT_F   128
#define NUM_K   16
#define M_COLS  (OUT_F * NUM_K)   // 2048
#define ROW_OUT (IN_F + OUT_F)    // 640

typedef __attribute__((ext_vector_type(2))) float v2f;
typedef __attribute__((ext_vector_type(8))) float v8f;

// ---------------------------------------------------------------------------
// Phase 1: m = x @ T  via V_WMMA_F32_16X16X4_F32 (fp32 in, fp32 accumulate).
// One wave computes a 16(M) x 128(N) strip: 8 accumulators (8 v8f = 64 VGPR),
// A-fragment loaded once per K-step and reused across 8 WMMAs.
// 256 wave-jobs total = 32 blocks x 8 waves.
// ---------------------------------------------------------------------------
__global__ void __launch_bounds__(256)
mbd_gemm_wmma_f32(const float* __restrict__ X,   // [256][512]
                  const float* __restrict__ T,   // [512][2048]
                  float* __restrict__ M)         // [256][2048]
{
    const int lane    = threadIdx.x & 31;
    const int wave    = threadIdx.x >> 5;
    const int job     = blockIdx.x * 8 + wave;     // 0..255
    const int mTile   = (job & 15) << 4;           // 0,16,...,240
    const int nBase   = (job >> 4) << 7;           // 0,128,...,1920
    const int halfSel = lane >> 4;                 // 0: lanes 0-15, 1: 16-31
    const int lrow    = lane & 15;

    const float* Arow = X + (size_t)(mTile + lrow) * IN_F;

    v8f acc[8] = {};

    for (int k0 = 0; k0 < IN_F; k0 += 4) {
        const int ka = k0 + (halfSel << 1);        // per-lane K base (0/2 split)

        // A fragment: 16x4 fp32, 2 VGPRs (consecutive K per lane -> b64 load)
        v2f a;
        a.x = Arow[ka];
        a.y = Arow[ka + 1];

        #pragma unroll
        for (int t = 0; t < 8; ++t) {
            const int col = nBase + (t << 4) + lrow;
            // B fragment: 4x16 fp32, 2 VGPRs, layout symmetric to A
            v2f b;
            b.x = T[(size_t)ka       * M_COLS + col];
            b.y = T[(size_t)(ka + 1) * M_COLS + col];
            // (neg_a, A, neg_b, B, c_mod, C, reuse_a, reuse_b)
            acc[t] = __builtin_amdgcn_wmma_f32_16x16x4_f32(
                         false, a, false, b, (short)0, acc[t], false, false);
        }
    }

    // D layout: acc[t][r] -> row mTile + 8*halfSel + r, col nBase + 16*t + lrow
    #pragma unroll
    for (int t = 0; t < 8; ++t) {
        const int col = nBase + (t << 4) + lrow;
        #pragma unroll
        for (int r = 0; r < 8; ++r) {
            M[(size_t)(mTile + (halfSel << 3) + r) * M_COLS + col] = acc[t][r];
        }
    }
}

// ---------------------------------------------------------------------------
// Phase 2: out[i,f] = sum_j exp(-sum_k |m[i,f,k]-m[j,f,k]|) - 1
// One block per feature f (128 blocks). Thread i keeps its 16 kernel values
// in VGPRs; the full 256x16 feature slice (16 KB) lives in LDS and the j-sweep
// reads it with wave-uniform (broadcast) ds_load_b128s.
// ---------------------------------------------------------------------------
__global__ void __launch_bounds__(256)
mbd_pairwise(const float* __restrict__ M,   // [256][2048]
             float* __restrict__ OUT)       // [256][640]
{
    __shared__ float sm[B_DIM * NUM_K];     // 16 KB

    const int f = blockIdx.x;               // 0..127
    const int i = threadIdx.x;              // 0..255

    float mi[NUM_K];
    const float* row = M + (size_t)i * M_COLS + f * NUM_K;
    #pragma unroll
    for (int k = 0; k < NUM_K; ++k) {
        mi[k] = row[k];
        sm[i * NUM_K + k] = mi[k];
    }
    __syncthreads();

    float total = 0.0f;
    for (int j = 0; j < B_DIM; ++j) {
        const float* mj = &sm[j * NUM_K];
        float s = 0.0f;
        #pragma unroll
        for (int k = 0; k < NUM_K; ++k)
            s += __builtin_fabsf(mi[k] - mj[k]);
        total += __expf(-s);                // j==i contributes exp(0)=1
    }

    OUT[(size_t)i * ROW_OUT + IN_F + f] = total - 1.0f;
}

// ---------------------------------------------------------------------------
// Phase 3: copy x into the first 512 columns of each 640-wide output row.
// float4 vectorized: 256 rows x 128 float4s = 32768 threads.
// ---------------------------------------------------------------------------
__global__ void __launch_bounds__(256)
mbd_copy_x(const float* __restrict__ X, float* __restrict__ OUT)
{
    const int idx = blockIdx.x * blockDim.x + threadIdx.x;  // 0..32767
    const int r = idx >> 7;        // row (128 float4 per row)
    const int c = idx & 127;
    const float4 v = ((const float4*)(X + (size_t)r * IN_F))[c];
    ((float4*)(OUT + (size_t)r * ROW_OUT))[c] = v;
}

extern "C" void kernel_launch(void* const* d_in, const int* in_sizes, int n_in,
                              void* d_out, int out_size, void* d_ws, size_t ws_size,
                              hipStream_t stream) {
    const float* X = (const float*)d_in[0];   // [256*512] f32
    const float* T = (const float*)d_in[1];   // [512*2048] f32
    float* OUT = (float*)d_out;               // [256*640] f32
    float* M   = (float*)d_ws;                // [256*2048] f32 = 2 MB scratch

    // Phase 1: WMMA GEMM. 256 wave-jobs, 8 waves/block -> 32 blocks.
    mbd_gemm_wmma_f32<<<dim3(32), dim3(256), 0, stream>>>(X, T, M);

    // Phase 2: pairwise L1 + exp reduction. One block per output feature.
    mbd_pairwise<<<dim3(OUT_F), dim3(256), 0, stream>>>(M, OUT);

    // Phase 3: concat x into output rows.
    mbd_copy_x<<<dim3(128), dim3(256), 0, stream>>>(X, OUT);
}